// AttentiveReadout_44040594653887
// MI455X (gfx1250) — compile-verified
//
#include <hip/hip_runtime.h>
#include <hip/hip_bf16.h>

typedef __attribute__((ext_vector_type(2))) float v2f;
typedef __attribute__((ext_vector_type(8))) float v8f;

#define DCOLS 256
#define TILE_ROWS 48          // 3 WMMA slabs of 16 rows
#define LDS_STRIDE 260        // pad 256->260 floats: conflict-free ds_load_b64 in both phases
#define THREADS 128           // 4 waves; waves 0-2 run WMMA slabs

__global__ void zero_out_kernel(float* __restrict__ out, int n) {
    int i = blockIdx.x * blockDim.x + threadIdx.x;
    if (i < n) out[i] = 0.0f;
}

__global__ __launch_bounds__(THREADS)
void attentive_readout_kernel(const float* __restrict__ x,
                              const long long* __restrict__ batch_id,
                              const float* __restrict__ gate_w,
                              const float* __restrict__ gate_b,
                              float* __restrict__ pooled,
                              int n_rows) {
    __shared__ float xlds[TILE_ROWS * LDS_STRIDE];   // 48*260*4 = 49,920 B
    __shared__ float glds[DCOLS];                    // gate weights
    __shared__ float wlds[TILE_ROWS];                // sigmoid gate per row
    __shared__ int   bidlds[TILE_ROWS];              // segment id per row

    const int t = threadIdx.x;
    const long long row0 = (long long)blockIdx.x * TILE_ROWS;

    // ---------- Phase 1: stage tile into LDS (coalesced b128), zero-fill tail ----------
    // 48 rows * 64 float4 = 3072 float4; 128 threads -> 24 each
    #pragma unroll 4
    for (int i = 0; i < 24; ++i) {
        int idx = i * THREADS + t;       // 0..3071
        int r   = idx >> 6;              // row in tile
        int c4  = idx & 63;              // float4 column
        long long grow = row0 + r;
        float4 v = make_float4(0.f, 0.f, 0.f, 0.f);
        if (grow < n_rows) {
            v = *reinterpret_cast<const float4*>(x + grow * DCOLS + (long long)c4 * 4);
        }
        // row*1040B is 16B-aligned (1040 = 65*16) -> ds_store_b128
        *reinterpret_cast<float4*>(&xlds[r * LDS_STRIDE + c4 * 4]) = v;
    }
    glds[t]         = gate_w[t];
    glds[t + 128]   = gate_w[t + 128];
    if (t < TILE_ROWS) {
        long long grow = row0 + t;
        long long id = (grow < n_rows) ? batch_id[grow] : batch_id[n_rows - 1];
        bidlds[t] = (int)id;             // out-of-range rows are zero -> contribute 0
    }
    __syncthreads();

    // ---------- Phase 2: gate dot via V_WMMA_F32_16X16X4_F32 (fp32-exact) ----------
    {
        const int wave = t >> 5;         // 0..3
        const int lane = t & 31;
        const int half = lane >> 4;      // K half select
        const int m    = lane & 15;      // row within slab
        if (wave < 3) {                  // uniform per wave; EXEC all-ones inside
            const int row = wave * 16 + m;
            v8f c = {0.f, 0.f, 0.f, 0.f, 0.f, 0.f, 0.f, 0.f};
            for (int j = 0; j < 64; ++j) {
                const int k = 4 * j + 2 * half;
                // A fragment: 16x4 slice of tile (lane layout per ISA 32-bit A 16x4)
                v2f a = *reinterpret_cast<const v2f*>(&xlds[row * LDS_STRIDE + k]);
                // B fragment: gate_w[k..k+1] broadcast to all 16 N-columns ->
                // identical K-indexing to A, broadcast LDS read (no conflicts)
                v2f b = *reinterpret_cast<const v2f*>(&glds[k]);
                c = __builtin_amdgcn_wmma_f32_16x16x4_f32(
                        /*neg_a=*/false, a, /*neg_b=*/false, b,
                        /*c_mod=*/(short)0, c, /*reuse_a=*/false, /*reuse_b=*/false);
            }
            // Every column of C equals the row dot. lanes 0-15 hold slab rows 0..7,
            // lanes 16-31 hold slab rows 8..15 (C layout: VGPR v -> M=v / M=v+8).
            if (m == 0) {
                const float bias = gate_b[0];
                #pragma unroll
                for (int v = 0; v < 8; ++v) {
                    float d = c[v] + bias;
                    float s = 1.0f / (1.0f + __expf(-d));
                    wlds[wave * 16 + half * 8 + v] = s;
                }
            }
        }
    }
    __syncthreads();

    // ---------- Phase 3: weighted segment-sum (sorted ids -> run-length flush) ----------
    {
        const int c0 = 2 * t;            // each thread owns 2 contiguous columns
        float acc0 = 0.f, acc1 = 0.f;
        int cur = bidlds[0];
        #pragma unroll 4
        for (int r = 0; r < TILE_ROWS; ++r) {
            int id = bidlds[r];
            if (id != cur) {
                atomicAdd(&pooled[(long long)cur * DCOLS + c0],     acc0);
                atomicAdd(&pooled[(long long)cur * DCOLS + c0 + 1], acc1);
                acc0 = 0.f; acc1 = 0.f;
                cur = id;
            }
            float w = wlds[r];
            acc0 += w * xlds[r * LDS_STRIDE + c0];
            acc1 += w * xlds[r * LDS_STRIDE + c0 + 1];
        }
        atomicAdd(&pooled[(long long)cur * DCOLS + c0],     acc0);
        atomicAdd(&pooled[(long long)cur * DCOLS + c0 + 1], acc1);
    }
}

extern "C" void kernel_launch(void* const* d_in, const int* in_sizes, int n_in,
                              void* d_out, int out_size, void* d_ws, size_t ws_size,
                              hipStream_t stream) {
    const float*     x        = (const float*)d_in[0];
    const long long* batch_id = (const long long*)d_in[1];
    // d_in[2] = batch_size scalar (encoded in out_size; unused)
    const float*     gate_w   = (const float*)d_in[3];
    const float*     gate_b   = (const float*)d_in[4];
    float*           pooled   = (float*)d_out;

    const int n_rows = in_sizes[0] / DCOLS;

    // Zero the accumulator output (harness poisons d_out); same stream -> ordered.
    {
        int zt = 256;
        int zb = (out_size + zt - 1) / zt;
        zero_out_kernel<<<zb, zt, 0, stream>>>(pooled, out_size);
    }

    int blocks = (n_rows + TILE_ROWS - 1) / TILE_ROWS;
    attentive_readout_kernel<<<blocks, THREADS, 0, stream>>>(
        x, batch_id, gate_w, gate_b, pooled, n_rows);
}